// PaiNNConv_25898652795338
// MI455X (gfx1250) — compile-verified
//
#include <hip/hip_runtime.h>
#include <hip/hip_bf16.h>

typedef __attribute__((ext_vector_type(2))) float v2f;
typedef __attribute__((ext_vector_type(8))) float v8f;

// D = A(16x4 f32) * B(4x16 f32) + C, fp32 WMMA (wave32).
__device__ __forceinline__ v8f wmma4(v2f a, v2f b, v8f c) {
    return __builtin_amdgcn_wmma_f32_16x16x4_f32(
        /*neg_a=*/false, a, /*neg_b=*/false, b,
        /*c_mod=*/(short)0, c, /*reuse_a=*/false, /*reuse_b=*/false);
}

__device__ __forceinline__ float silu(float x) {
    return x / (1.0f + __expf(-x));
}

// -------------------------------------------------------------------------
// Kernel A: s_full[m, 0:192] = silu(node @ Ws^T + bs) @ Wphi^T + bphi
// One wave per 16-row tile. 8 waves / block.
// -------------------------------------------------------------------------
__global__ __launch_bounds__(256) void painn_node_mlp(
    const float* __restrict__ node, const float* __restrict__ Ws,
    const float* __restrict__ bs, const float* __restrict__ Wphi,
    const float* __restrict__ bphi, float* __restrict__ s_full, int n_tiles) {
    __shared__ float s1[8][16][68];   // 68-float row stride: conflict-free transposed reads

    const int wave = threadIdx.x >> 5;
    const int lane = threadIdx.x & 31;
    const int tile = blockIdx.x * 8 + wave;
    if (tile >= n_tiles) return;

    const int m0   = tile * 16;
    const int mrow = lane & 15;    // A row / C-D column for this lane
    const int half = lane >> 4;    // 0: K=k0..k0+1 / M=vgpr, 1: K=k0+2..k0+3 / M=vgpr+8
    const int koff = half * 2;

    // Preload node-tile A fragments: 16 K-steps of 4.
    v2f a[16];
    const float* nrow = node + (size_t)(m0 + mrow) * 64;
#pragma unroll
    for (int kk = 0; kk < 16; ++kk)
        a[kk] = *(const v2f*)(nrow + kk * 4 + koff);

    float* s1w = &s1[wave][0][0];

    // GEMM1: (16x64) @ (64x64)^T, bias + SiLU, stage to LDS.
#pragma unroll
    for (int t = 0; t < 4; ++t) {
        const int n0 = t * 16;
        v8f c = {};
#pragma unroll
        for (int kk = 0; kk < 16; ++kk) {
            v2f b = *(const v2f*)(Ws + (size_t)(n0 + mrow) * 64 + kk * 4 + koff);
            c = wmma4(a[kk], b, c);
        }
        const float bias = bs[n0 + mrow];
#pragma unroll
        for (int i = 0; i < 8; ++i) {
            const int r = i + half * 8;                 // C/D row
            s1w[r * 68 + n0 + mrow] = silu(c[i] + bias);
        }
    }

    // Wave-private LDS: wait for our own DS stores before transposed reload.
    asm volatile("s_wait_dscnt 0" ::: "memory");

    // Reload staged s1 as A fragments.
    v2f a2[16];
#pragma unroll
    for (int kk = 0; kk < 16; ++kk)
        a2[kk] = *(const v2f*)(s1w + mrow * 68 + kk * 4 + koff);

    // GEMM2: (16x64) @ (64x192)^T + bphi -> s_full.
#pragma unroll
    for (int t = 0; t < 12; ++t) {
        const int n0 = t * 16;
        v8f c = {};
#pragma unroll
        for (int kk = 0; kk < 16; ++kk) {
            v2f b = *(const v2f*)(Wphi + (size_t)(n0 + mrow) * 64 + kk * 4 + koff);
            c = wmma4(a2[kk], b, c);
        }
        const float bias = bphi[n0 + mrow];
#pragma unroll
        for (int i = 0; i < 8; ++i)
            s_full[(size_t)(m0 + i + half * 8) * 192 + n0 + mrow] = c[i] + bias;
    }
}

// -------------------------------------------------------------------------
// Kernel B: per 16-edge tile: w = (rbf @ Ww^T + bw)*env, sw = s_e * w,
// scatter-add ds (chunk 1) and dv (chunks 2+3 fused -> single atomic each).
// -------------------------------------------------------------------------
__global__ __launch_bounds__(256) void painn_edge(
    const float* __restrict__ rbf, const float* __restrict__ env,
    const float* __restrict__ rij, const int* __restrict__ edge_index,
    const float* __restrict__ Ww, const float* __restrict__ bw,
    const float* __restrict__ s_full, const float* __restrict__ equiv,
    float* __restrict__ ds_out, float* __restrict__ dv_out, int n_tiles) {
    __shared__ int   sDst[8][16];
    __shared__ int   sSrc[8][16];
    __shared__ float sEnv[8][16];
    __shared__ float sRij[8][16][3];

    const int wave = threadIdx.x >> 5;
    const int lane = threadIdx.x & 31;
    const int tile = blockIdx.x * 8 + wave;
    if (tile >= n_tiles) return;

    const int e0   = tile * 16;
    const int mrow = lane & 15;
    const int half = lane >> 4;
    const int koff = half * 2;

    if (lane < 16) {
        const int e = e0 + mrow;
        sDst[wave][mrow]    = edge_index[2 * (size_t)e + 0];
        sSrc[wave][mrow]    = edge_index[2 * (size_t)e + 1];
        sEnv[wave][mrow]    = env[e];
        sRij[wave][mrow][0] = rij[3 * (size_t)e + 0];
        sRij[wave][mrow][1] = rij[3 * (size_t)e + 1];
        sRij[wave][mrow][2] = rij[3 * (size_t)e + 2];
    }
    // Wave-private staging: only our own DS stores need to land.
    asm volatile("s_wait_dscnt 0" ::: "memory");

    // rbf tile A fragments: K=20 -> 5 K-steps of 4.
    v2f a[5];
    const float* rrow = rbf + (size_t)(e0 + mrow) * 20;
#pragma unroll
    for (int kk = 0; kk < 5; ++kk)
        a[kk] = *(const v2f*)(rrow + kk * 4 + koff);

    // ---- ds chunk: output columns 0..63 (n-tiles 0..3) ----
#pragma unroll
    for (int t = 0; t < 4; ++t) {
        const int n0 = t * 16;
        v8f c = {};
#pragma unroll
        for (int kk = 0; kk < 5; ++kk) {
            v2f b = *(const v2f*)(Ww + (size_t)(n0 + mrow) * 20 + kk * 4 + koff);
            c = wmma4(a[kk], b, c);
        }
        const float bias = bw[n0 + mrow];
#pragma unroll
        for (int i = 0; i < 8; ++i) {
            const int   me = i + half * 8;          // edge within tile
            const float w  = (c[i] + bias) * sEnv[wave][me];
            const float se = s_full[(size_t)sSrc[wave][me] * 192 + n0 + mrow];
            unsafeAtomicAdd(&ds_out[(size_t)sDst[wave][me] * 64 + n0 + mrow], w * se);
        }
    }

    // ---- dv chunks: columns 64..127 (sw2) paired with 128..191 (sw3) ----
#pragma unroll
    for (int t = 0; t < 4; ++t) {
        const int n2 = 64 + t * 16;
        const int n3 = 128 + t * 16;
        v8f c2 = {}, c3 = {};
#pragma unroll
        for (int kk = 0; kk < 5; ++kk) {
            v2f b2 = *(const v2f*)(Ww + (size_t)(n2 + mrow) * 20 + kk * 4 + koff);
            c2 = wmma4(a[kk], b2, c2);
        }
#pragma unroll
        for (int kk = 0; kk < 5; ++kk) {
            v2f b3 = *(const v2f*)(Ww + (size_t)(n3 + mrow) * 20 + kk * 4 + koff);
            c3 = wmma4(a[kk], b3, c3);
        }
        const float bias2 = bw[n2 + mrow];
        const float bias3 = bw[n3 + mrow];
        const int   nn    = t * 16 + mrow;          // column within the 64-wide chunk
#pragma unroll
        for (int i = 0; i < 8; ++i) {
            const int   me  = i + half * 8;
            const int   src = sSrc[wave][me];
            const int   dst = sDst[wave][me];
            const float ee  = sEnv[wave][me];
            const float sw2 = (c2[i] + bias2) * ee * s_full[(size_t)src * 192 + n2 + mrow];
            const float sw3 = (c3[i] + bias3) * ee * s_full[(size_t)src * 192 + n3 + mrow];
#pragma unroll
            for (int d = 0; d < 3; ++d) {
                const float vj      = equiv[(size_t)src * 192 + d * 64 + nn];
                const float contrib = vj * sw2 + sRij[wave][me][d] * sw3;
                unsafeAtomicAdd(&dv_out[(size_t)dst * 192 + d * 64 + nn], contrib);
            }
        }
    }
}

extern "C" void kernel_launch(void* const* d_in, const int* in_sizes, int n_in,
                              void* d_out, int out_size, void* d_ws, size_t ws_size,
                              hipStream_t stream) {
    const float* node  = (const float*)d_in[0];
    const float* equiv = (const float*)d_in[1];
    const float* rbf   = (const float*)d_in[2];
    const float* env   = (const float*)d_in[3];
    const float* rij   = (const float*)d_in[4];
    const int*   eidx  = (const int*)d_in[5];
    const float* Ws    = (const float*)d_in[6];
    const float* bs    = (const float*)d_in[7];
    const float* Wphi  = (const float*)d_in[8];
    const float* bphi  = (const float*)d_in[9];
    const float* Ww    = (const float*)d_in[10];
    const float* bw    = (const float*)d_in[11];

    const int n_nodes = in_sizes[0] / 64;   // 50000
    const int n_edges = in_sizes[3];        // envelope has one element per edge

    float* s_full = (float*)d_ws;                         // n_nodes * 192 f32
    float* ds_out = (float*)d_out;                        // n_nodes * 64
    float* dv_out = ds_out + (size_t)n_nodes * 64;        // n_nodes * 3 * 64

    // Outputs are accumulated with atomics: zero them every call (capture-safe).
    hipMemsetAsync(d_out, 0, (size_t)out_size * sizeof(float), stream);

    const int node_tiles = (n_nodes + 15) / 16;           // 3125 (exact)
    painn_node_mlp<<<(node_tiles + 7) / 8, 256, 0, stream>>>(
        node, Ws, bs, Wphi, bphi, s_full, node_tiles);

    const int edge_tiles = (n_edges + 15) / 16;           // 50000 (exact)
    painn_edge<<<(edge_tiles + 7) / 8, 256, 0, stream>>>(
        rbf, env, rij, eidx, Ww, bw, s_full, equiv, ds_out, dv_out, edge_tiles);
}